// GraphMertGraphNodeFeature_22024592294142
// MI455X (gfx1250) — compile-verified
//
#include <hip/hip_runtime.h>

#define H         128
#define ROOT      128
#define MAX_NODES 1024
#define NL        7           // leaves per root
#define LH        4           // head span
#define BATCH     64
#define NOUT      (MAX_NODES + 1)   // 1025 rows per batch in output

typedef __attribute__((ext_vector_type(2))) float v2f;
typedef __attribute__((ext_vector_type(8))) float v8f;

// ---------------------------------------------------------------------------
// Kernel 1: out[b,0,:] = graph_token ; out[b,1+n,:] = atom_emb[tok[b,n]]
// Fully coalesced: 1 float per thread.
// ---------------------------------------------------------------------------
__global__ __launch_bounds__(256) void base_fill_kernel(
    const int*   __restrict__ input_nodes,   // [B,1024]
    const float* __restrict__ atom_emb,      // [30000,128]
    const float* __restrict__ graph_token,   // [128]
    float*       __restrict__ out)           // [B,1025,128]
{
    int idx = blockIdx.x * blockDim.x + threadIdx.x;
    const int total = BATCH * NOUT * H;
    if (idx >= total) return;
    int j    = idx & (H - 1);
    int row  = idx / H;            // b*1025 + node
    int node = row % NOUT;
    int b    = row / NOUT;
    float v;
    if (node == 0) {
        v = graph_token[j];
    } else {
        int tok = input_nodes[b * MAX_NODES + (node - 1)];
        v = atom_emb[(size_t)tok * H + j];
    }
    out[idx] = v;
}

// ---------------------------------------------------------------------------
// Kernel 2: one 256-thread (8-wave) workgroup per triple s = b*128 + root.
//   v_a = M^T r_even, v_b = M^T r_odd        (coalesced column sweeps of M)
//   a[t] = v_a . te[t], b[h] = v_b . he[h]   (wave shuffle reductions)
//   alpha = softmax_h(leaky_relu(a+b)); hsum[t] = sum_h alpha * he[h]
//   final = M(128x128) x hsum^T(128x16pad)   (V_WMMA_F32_16X16X4_F32, fp32)
//   out[b, 129 + root*7 + t, :] += final[:,t]  (disjoint per block -> no atomics)
// ---------------------------------------------------------------------------
__global__ __launch_bounds__(256) void triple_kernel(
    const int*   __restrict__ input_nodes,   // [B,1024]
    const int*   __restrict__ leaf_rel,      // [B,128]
    const float* __restrict__ atom_emb,      // [30000,128]
    const float* __restrict__ rel_mat_emb,   // [65,128*128]
    const float* __restrict__ rel_emb,       // [65,256]
    float*       __restrict__ out)           // [B,1025,128]
{
    __shared__ float s_he[LH][H];
    __shared__ float s_te[NL][H];
    __shared__ float s_va[H];
    __shared__ float s_vb[H];
    __shared__ float s_r[2 * H];
    __shared__ float s_hsum[16][130];        // hsum^T staged row t, 130-stride (conflict-free b64)
    __shared__ float s_ab[16];               // [0..6]=a[t], [7..10]=b[h]

    const int s    = blockIdx.x;
    const int b    = s >> 7;
    const int root = s & 127;
    const int tid  = threadIdx.x;
    const int lane = tid & 31;
    const int wave = tid >> 5;

    const int rel = leaf_rel[b * ROOT + root];
    const float* __restrict__ M = rel_mat_emb + (size_t)rel * (H * H);

    // warm M (64KB) into caches: global_prefetch_b8, 256B stride covers the matrix
    __builtin_prefetch(M + tid * 64, 0, 0);

    // relation vector r -> LDS (256 floats)
    s_r[tid] = rel_emb[(size_t)rel * (2 * H) + tid];

    // gather head/tail embeddings into LDS
    for (int idx = tid; idx < LH * H; idx += 256) {
        int h = idx >> 7, j = idx & 127;
        int hp = root + h; if (hp > ROOT - 1) hp = ROOT - 1;    // jnp.clip
        int tok = input_nodes[b * MAX_NODES + hp];
        s_he[h][j] = atom_emb[(size_t)tok * H + j];
    }
    for (int idx = tid; idx < NL * H; idx += 256) {
        int t = idx / H, j = idx & 127;
        int node = ROOT + root * NL + t;
        int tok = input_nodes[b * MAX_NODES + node];
        s_te[t][j] = atom_emb[(size_t)tok * H + j];
    }
    // zero hsum buffer (pad rows 7..15 must stay zero for the WMMA B operand)
    for (int idx = tid; idx < 16 * 130; idx += 256)
        ((float*)s_hsum)[idx] = 0.0f;
    __syncthreads();

    // phase 1: v_a[j] = sum_i r[2i]*M[i][j] ; v_b[j] = sum_i r[2i+1]*M[i][j]
    {
        int j    = tid & 127;
        int half = tid >> 7;                 // 0 -> even r (tails), 1 -> odd r (heads)
        float acc = 0.0f;
        #pragma unroll 8
        for (int i = 0; i < H; ++i)
            acc = fmaf(s_r[2 * i + half], M[i * H + j], acc);   // coalesced across j
        if (half == 0) s_va[j] = acc; else s_vb[j] = acc;
    }
    __syncthreads();

    // phase 2: 11 dot products of length 128, one wave each (looped)
    for (int item = wave; item < NL + LH; item += 8) {
        const float* vp = (item < NL) ? s_va : s_vb;
        const float* xp = (item < NL) ? s_te[item] : s_he[item - NL];
        float p = 0.0f;
        #pragma unroll 4
        for (int j = lane; j < H; j += 32)
            p = fmaf(vp[j], xp[j], p);
        #pragma unroll
        for (int off = 16; off > 0; off >>= 1)
            p += __shfl_xor(p, off, 32);
        if (lane == 0) s_ab[item] = p;
    }
    __syncthreads();

    // phase 3: leaky-relu + softmax over 4 heads, hsum[t][j] into LDS
    if (tid < H) {
        int j = tid;
        #pragma unroll
        for (int t = 0; t < NL; ++t) {
            float sc[LH], mx = -1e30f;
            #pragma unroll
            for (int h = 0; h < LH; ++h) {
                float x = s_ab[t] + s_ab[NL + h];
                x = (x > 0.0f) ? x : 0.01f * x;          // leaky_relu(0.01)
                sc[h] = x;
                mx = fmaxf(mx, x);
            }
            float sum = 0.0f;
            #pragma unroll
            for (int h = 0; h < LH; ++h) { sc[h] = __expf(sc[h] - mx); sum += sc[h]; }
            float inv = 1.0f / sum;
            float acc = 0.0f;
            #pragma unroll
            for (int h = 0; h < LH; ++h)
                acc = fmaf(sc[h] * inv, s_he[h][j], acc);
            s_hsum[t][j] = acc;
        }
    }
    __syncthreads();

    // phase 4: final = M x hsum^T via chained V_WMMA_F32_16X16X4_F32.
    // Wave w owns output rows i in [16w,16w+16). 32 K-steps of 4 accumulate K=128.
    //   A 16x4 f32 frag: lanes 0-15 row m=lane, K={4kb,4kb+1}; lanes 16-31 K={4kb+2,4kb+3}
    //   B 4x16 f32 frag: lanes 0-15 col n=lane, K={4kb,4kb+1}; lanes 16-31 K={4kb+2,4kb+3}
    {
        const int lane_m  = lane & 15;
        const int lane_hi = lane >> 4;
        const int i0      = wave * 16;
        const float* Arow = M + (size_t)(i0 + lane_m) * H + 2 * lane_hi;    // 8B-aligned pairs
        const float* Brow = &s_hsum[lane_m][2 * lane_hi];                   // 130-stride, 8B-aligned
        v8f c = {};
        #pragma unroll
        for (int kb = 0; kb < 32; ++kb) {
            v2f af = *reinterpret_cast<const v2f*>(Arow + 4 * kb);
            v2f bf = *reinterpret_cast<const v2f*>(Brow + 4 * kb);
            c = __builtin_amdgcn_wmma_f32_16x16x4_f32(
                    /*neg_a=*/false, af, /*neg_b=*/false, bf,
                    /*c_mod=*/(short)0, c, /*reuse_a=*/false, /*reuse_b=*/false);
        }
        // D layout: VGPR v, lane: i = i0 + v + 8*lane_hi, t = lane_m. Scatter-add leaves.
        int t = lane_m;
        if (t < NL) {
            size_t base = ((size_t)(b * NOUT) + 1 + ROOT + root * NL + t) * H;
            #pragma unroll
            for (int vv = 0; vv < 8; ++vv) {
                int i = i0 + vv + 8 * lane_hi;
                out[base + i] += c[vv];      // disjoint across blocks/lanes; ordered after kernel 1
            }
        }
    }
}

// ---------------------------------------------------------------------------
extern "C" void kernel_launch(void* const* d_in, const int* in_sizes, int n_in,
                              void* d_out, int out_size, void* d_ws, size_t ws_size,
                              hipStream_t stream) {
    const int*   input_nodes = (const int*)  d_in[0];   // [B,1024,1] int32
    const int*   leaf_rel    = (const int*)  d_in[1];   // [B,128]    int32
    // d_in[2] = head_lengths (all == LH, clamp handled analytically) -- unused
    const float* atom_emb    = (const float*)d_in[3];   // [30000,128]
    const float* rel_mat_emb = (const float*)d_in[4];   // [65,16384]
    const float* rel_emb     = (const float*)d_in[5];   // [65,256]
    const float* graph_token = (const float*)d_in[6];   // [1,128]
    float* out = (float*)d_out;                         // [B,1025,128]

    const int total = BATCH * NOUT * H;
    base_fill_kernel<<<(total + 255) / 256, 256, 0, stream>>>(
        input_nodes, atom_emb, graph_token, out);
    triple_kernel<<<BATCH * ROOT, 256, 0, stream>>>(
        input_nodes, leaf_rel, atom_emb, rel_mat_emb, rel_emb, out);
}